// CustomSAGE_APPNP_45466523795733
// MI455X (gfx1250) — compile-verified
//
#include <hip/hip_runtime.h>

typedef __attribute__((ext_vector_type(2))) float v2f;
typedef __attribute__((ext_vector_type(8))) float v8f;

#define HIDC 128   // IN == HID == 128 in the reference

// ---------------------------------------------------------------- utilities
__global__ void k_zero(float* __restrict__ p, long n) {
  long i = (long)blockIdx.x * blockDim.x + threadIdx.x;
  long stride = (long)gridDim.x * blockDim.x;
  for (; i < n; i += stride) p[i] = 0.0f;
}

// ------------------------------------------------- SAGE neighbor aggregation
// wave(32 lanes) per edge: lane handles 4 channels (float4), atomic add to agg[dst]
__global__ void k_edge_agg(const float* __restrict__ x, const int* __restrict__ src,
                           const int* __restrict__ dst, float* __restrict__ agg,
                           float* __restrict__ cnt, float* __restrict__ deg, int E) {
  long tid = (long)blockIdx.x * blockDim.x + threadIdx.x;
  int e = (int)(tid >> 5);
  int lane = (int)(tid & 31);
  if (e >= E) return;
  int s = src[e], d = dst[e];
  const float4 v = *(const float4*)(x + (long)s * HIDC + lane * 4);
  float* a = agg + (long)d * HIDC + lane * 4;
  atomicAdd(a + 0, v.x);
  atomicAdd(a + 1, v.y);
  atomicAdd(a + 2, v.z);
  atomicAdd(a + 3, v.w);
  if (lane == 0) {
    atomicAdd(cnt + d, 1.0f);
    atomicAdd(deg + d, 1.0f);
  }
}

// agg /= max(cnt,1)  (elementwise over N*128)
__global__ void k_mean(float* __restrict__ agg, const float* __restrict__ cnt, long total) {
  long i = (long)blockIdx.x * blockDim.x + threadIdx.x;
  long stride = (long)gridDim.x * blockDim.x;
  for (; i < total; i += stride) {
    float c = fmaxf(cnt[i >> 7], 1.0f);
    agg[i] = agg[i] / c;
  }
}

// dinv = rsqrt(indeg + 1)  (self-loop included; deg >= 1 always)
__global__ void k_dinv(const float* __restrict__ deg, float* __restrict__ dinv, int N) {
  int i = blockIdx.x * blockDim.x + threadIdx.x;
  if (i < N) dinv[i] = rsqrtf(deg[i] + 1.0f);
}

// per-edge gcn norm, pre-scaled by (1-ALPHA)=0.9: norm = 0.9*dinv[src]*dinv[dst]
__global__ void k_norm(const int* __restrict__ src, const int* __restrict__ dst,
                       const float* __restrict__ dinv, float* __restrict__ norm, int E) {
  int e = blockIdx.x * blockDim.x + threadIdx.x;
  if (e < E) norm[e] = 0.9f * dinv[src[e]] * dinv[dst[e]];
}

// ------------------------------------------------------------- SAGE GEMM
// hpre = mean_nb @ W_l + x @ W_r + b_sage using V_WMMA_F32_16X16X4_F32.
// Block: 256 threads = 8 waves; block handles 16 rows (nodes) x 128 cols;
// each wave owns one 16x16 output tile, K-loop of 64 WMMA ops (2 matrices x 32).
// A frag (16x4 f32): lanes 0-15 -> K = {k0,k0+1}, lanes 16-31 -> K = {k0+2,k0+3}.
// B frag (4x16 f32): same K split across lane halves, col = lane%16.
// C/D (16x16 f32):   VGPR v, lanes 0-15 -> M=v, lanes 16-31 -> M=v+8, col = lane%16.
__global__ __launch_bounds__(256)
void k_sage_wmma(const float* __restrict__ mean_nb, const float* __restrict__ x,
                 const float* __restrict__ Wl, const float* __restrict__ Wr,
                 const float* __restrict__ bias, float* __restrict__ hpre, int N) {
  const int wave = threadIdx.x >> 5;     // 0..7 -> column tile
  const int lane = threadIdx.x & 31;
  const int half = lane >> 4;            // 0 or 1
  const int l16  = lane & 15;
  const int row0 = blockIdx.x * 16;
  const int c0   = wave * 16;
  const int arow_idx = min(row0 + l16, N - 1);   // clamp keeps EXEC all-1s

  v8f acc = {};

  const float* __restrict__ ap = mean_nb + (long)arow_idx * HIDC + 2 * half;
  const float* __restrict__ bp = Wl + (long)(2 * half) * HIDC + c0 + l16;
#pragma unroll
  for (int k0 = 0; k0 < HIDC; k0 += 4) {
    v2f a; a.x = ap[k0];        a.y = ap[k0 + 1];
    v2f b; b.x = bp[k0 * HIDC]; b.y = bp[k0 * HIDC + HIDC];
    acc = __builtin_amdgcn_wmma_f32_16x16x4_f32(false, a, false, b,
                                                (short)0, acc, false, false);
  }
  ap = x + (long)arow_idx * HIDC + 2 * half;
  bp = Wr + (long)(2 * half) * HIDC + c0 + l16;
#pragma unroll
  for (int k0 = 0; k0 < HIDC; k0 += 4) {
    v2f a; a.x = ap[k0];        a.y = ap[k0 + 1];
    v2f b; b.x = bp[k0 * HIDC]; b.y = bp[k0 * HIDC + HIDC];
    acc = __builtin_amdgcn_wmma_f32_16x16x4_f32(false, a, false, b,
                                                (short)0, acc, false, false);
  }

  const float bv = bias[c0 + l16];
#pragma unroll
  for (int v = 0; v < 8; ++v) {
    int row = row0 + v + 8 * half;
    if (row < N) hpre[(long)row * HIDC + c0 + l16] = acc[v] + bv;
  }
}

// ---------------------------------------------------------- BatchNorm stats
// blockDim = 128 (one thread per channel); rows strided by gridDim
__global__ __launch_bounds__(128)
void k_stats(const float* __restrict__ h, float* __restrict__ stats, int N) {
  const int c = threadIdx.x;
  float s = 0.0f, s2 = 0.0f;
  for (int r = blockIdx.x; r < N; r += gridDim.x) {
    float v = h[(long)r * HIDC + c];
    s += v; s2 += v * v;
  }
  atomicAdd(stats + c, s);
  atomicAdd(stats + HIDC + c, s2);
}

__global__ void k_bn_relu(const float* __restrict__ hpre, const float* __restrict__ stats,
                          const float* __restrict__ gamma, const float* __restrict__ beta,
                          float* __restrict__ h0, long total, float invN) {
  long i = (long)blockIdx.x * blockDim.x + threadIdx.x;
  long stride = (long)gridDim.x * blockDim.x;
  for (; i < total; i += stride) {
    int c = (int)(i & (HIDC - 1));
    float mu  = stats[c] * invN;
    float var = stats[HIDC + c] * invN - mu * mu;
    float v = (hpre[i] - mu) * rsqrtf(var + 1e-5f) * gamma[c] + beta[c];
    h0[i] = v > 0.0f ? v : 0.0f;
  }
}

// -------------------------------------------------------- APPNP propagation
// Fused pre-init: tgt = 0.9*dinv^2*cur + 0.1*h0   (self-loop + teleport term),
// so the edge scatter (with 0.9-prescaled norms) completes the iteration with
// no separate zero or combine pass. float4-vectorized; total4 = N*128/4.
__global__ void k_prop_init(float4* __restrict__ tgt, const float4* __restrict__ cur,
                            const float4* __restrict__ h0, const float* __restrict__ dinv,
                            long total4) {
  long i = (long)blockIdx.x * blockDim.x + threadIdx.x;
  long stride = (long)gridDim.x * blockDim.x;
  for (; i < total4; i += stride) {
    float a = dinv[i >> 5];          // 32 float4 per node row
    float s = 0.9f * a * a;
    float4 c = cur[i];
    float4 z = h0[i];
    float4 r;
    r.x = s * c.x + 0.1f * z.x;
    r.y = s * c.y + 0.1f * z.y;
    r.z = s * c.z + 0.1f * z.z;
    r.w = s * c.w + 0.1f * z.w;
    tgt[i] = r;
  }
}

__global__ void k_prop_scatter(const float* __restrict__ cur, const int* __restrict__ src,
                               const int* __restrict__ dst, const float* __restrict__ norm,
                               float* __restrict__ tgt, int E) {
  long tid = (long)blockIdx.x * blockDim.x + threadIdx.x;
  int e = (int)(tid >> 5);
  int lane = (int)(tid & 31);
  if (e >= E) return;
  int s = src[e], d = dst[e];
  float w = norm[e];                 // already scaled by 0.9
  const float4 v = *(const float4*)(cur + (long)s * HIDC + lane * 4);
  float* t = tgt + (long)d * HIDC + lane * 4;
  atomicAdd(t + 0, w * v.x);
  atomicAdd(t + 1, w * v.y);
  atomicAdd(t + 2, w * v.z);
  atomicAdd(t + 3, w * v.w);
}

// ------------------------------------------------------------------ pooling
__global__ void k_pool(const float* __restrict__ h, const int* __restrict__ batch,
                       float* __restrict__ gsum, float* __restrict__ gcnt, int N) {
  long tid = (long)blockIdx.x * blockDim.x + threadIdx.x;
  int i = (int)(tid >> 5);
  int lane = (int)(tid & 31);
  if (i >= N) return;
  int g = batch[i];
  const float4 v = *(const float4*)(h + (long)i * HIDC + lane * 4);
  float* t = gsum + (long)g * HIDC + lane * 4;
  atomicAdd(t + 0, v.x);
  atomicAdd(t + 1, v.y);
  atomicAdd(t + 2, v.z);
  atomicAdd(t + 3, v.w);
  if (lane == 0) atomicAdd(gcnt + g, 1.0f);
}

__global__ void k_final(const float* __restrict__ gsum, const float* __restrict__ gcnt,
                        const float* __restrict__ Wlin, const float* __restrict__ blin,
                        float* __restrict__ out, int G) {
  int g = blockIdx.x * blockDim.x + threadIdx.x;
  if (g >= G) return;
  float inv = 1.0f / fmaxf(gcnt[g], 1.0f);
  float a0 = blin[0], a1 = blin[1];
  for (int c = 0; c < HIDC; ++c) {
    float p = gsum[(long)g * HIDC + c] * inv;
    a0 += p * Wlin[c * 2 + 0];
    a1 += p * Wlin[c * 2 + 1];
  }
  out[g * 2 + 0] = a0;
  out[g * 2 + 1] = a1;
}

// ------------------------------------------------------------------ launcher
extern "C" void kernel_launch(void* const* d_in, const int* in_sizes, int n_in,
                              void* d_out, int out_size, void* d_ws, size_t ws_size,
                              hipStream_t stream) {
  const float* x     = (const float*)d_in[0];
  const int*   ei    = (const int*)d_in[1];
  const int*   batch = (const int*)d_in[2];
  const float* Wl    = (const float*)d_in[3];
  const float* Wr    = (const float*)d_in[4];
  const float* bs    = (const float*)d_in[5];
  const float* gamma = (const float*)d_in[6];
  const float* beta  = (const float*)d_in[7];
  const float* Wlin  = (const float*)d_in[8];
  const float* blin  = (const float*)d_in[9];
  float* out = (float*)d_out;

  const int N = in_sizes[2];
  const int E = in_sizes[1] / 2;
  const int G = out_size / 2;
  const int* src = ei;
  const int* dst = ei + E;

  const long NH = (long)N * HIDC;
  float* ws   = (float*)d_ws;
  float* agg  = ws;              // N*128; neighbor mean, later APPNP pong buffer
  float* h0   = ws + NH;         // N*128; fixed teleport state
  float* buf  = ws + 2 * NH;     // N*128; hpre, later APPNP ping buffer
  float* cnt  = ws + 3 * NH;     // N
  float* deg  = cnt + N;         // N
  float* dinv = deg + N;         // N
  float* norm = dinv + N;        // E
  float* stats = norm + E;       // 256 (sum, sumsq)
  float* gsum  = stats + 256;    // G*128
  float* gcnt  = gsum + (long)G * HIDC; // G

  const long tailN = 3L * N + E + 256 + (long)G * HIDC + G;

  // zero accumulation buffers
  k_zero<<<1024, 256, 0, stream>>>(agg, NH);
  k_zero<<<512, 256, 0, stream>>>(cnt, tailN);

  // SAGE aggregation + degrees
  long eth = (long)E * 32;
  k_edge_agg<<<(int)((eth + 255) / 256), 256, 0, stream>>>(x, src, dst, agg, cnt, deg, E);
  k_mean<<<1024, 256, 0, stream>>>(agg, cnt, NH);
  k_dinv<<<(N + 255) / 256, 256, 0, stream>>>(deg, dinv, N);
  k_norm<<<(E + 255) / 256, 256, 0, stream>>>(src, dst, dinv, norm, E);

  // SAGE GEMM (WMMA f32): buf = mean@Wl + x@Wr + b
  k_sage_wmma<<<(N + 15) / 16, 256, 0, stream>>>(agg, x, Wl, Wr, bs, buf, N);

  // BatchNorm + ReLU -> h0
  k_stats<<<512, 128, 0, stream>>>(buf, stats, N);
  k_bn_relu<<<2048, 256, 0, stream>>>(buf, stats, gamma, beta, h0, NH, 1.0f / (float)N);

  // APPNP: K=10 iterations, ping-pong between buf and agg, h0 untouched.
  // Each iteration: fused init (self-loop + teleport) then edge scatter.
  float* cur = h0;
  float* tgt = buf;
  const long NH4 = NH / 4;
  for (int it = 0; it < 10; ++it) {
    k_prop_init<<<1024, 256, 0, stream>>>((float4*)tgt, (const float4*)cur,
                                          (const float4*)h0, dinv, NH4);
    k_prop_scatter<<<(int)((eth + 255) / 256), 256, 0, stream>>>(cur, src, dst, norm, tgt, E);
    cur = tgt;
    tgt = (tgt == buf) ? agg : buf;
  }

  // global mean pool + linear head
  long pth = (long)N * 32;
  k_pool<<<(int)((pth + 255) / 256), 256, 0, stream>>>(cur, batch, gsum, gcnt, N);
  k_final<<<(G + 63) / 64, 64, 0, stream>>>(gsum, gcnt, Wlin, blin, out, G);
}